// BezierGlyph_42949672960311
// MI455X (gfx1250) — compile-verified
//
#include <hip/hip_runtime.h>

#define IMG_SIZE   512
#define N_SAMP_TOT 256                       // 8 strokes * 32 samples
#define PIX_PER_WAVE 16
#define WAVES_PER_BLOCK 8
#define THREADS   (WAVES_PER_BLOCK * 32)
#define N_TILES   (N_SAMP_TOT / 16)          // 16

// exp(-256*dist) == exp2(-sqrt(dist^2 * K)),  K = (256*log2(e))^2.
// K is pre-baked into the B (sample) matrix so WMMA emits dist^2*K directly.
#define SHARP_L2E_SQ  136404.59749446486f

typedef __attribute__((ext_vector_type(2))) float v2f;
typedef __attribute__((ext_vector_type(8))) float v8f;

__device__ __forceinline__ float clamp01(float x) {
    return fminf(fmaxf(x, 0.0f), 1.0f);
}

__global__ __launch_bounds__(THREADS)
void bezier_glyph_wmma(const float* __restrict__ cp,          // (8,4,2)
                       const float* __restrict__ pixel_grid,  // (262144,2)
                       float* __restrict__ out)               // (262144)
{
    // LDS sample tables, pre-baked as K-scaled WMMA B-operand halves:
    //   sLo[n] = {-2*sx*K, -2*sy*K}   (K-rows 0,1 for lanes 0..15)
    //   sHi[n] = {     K , |s|^2*K}   (K-rows 2,3 for lanes 16..31)
    __shared__ float2 sLo[N_SAMP_TOT];
    __shared__ float2 sHi[N_SAMP_TOT];

    const int tid = threadIdx.x;

    // ---- Phase 1: every block evaluates all 256 Bezier samples into LDS ----
    {
        const int k = tid >> 5;          // stroke 0..7
        const int j = tid & 31;          // sample-on-stroke 0..31
        const float* c = cp + k * 8;     // 4 control points x,y
        const float p0x = clamp01(c[0]), p0y = clamp01(c[1]);
        const float p1x = clamp01(c[2]), p1y = clamp01(c[3]);
        const float p2x = clamp01(c[4]), p2y = clamp01(c[5]);
        const float p3x = clamp01(c[6]), p3y = clamp01(c[7]);
        const float t  = (float)j * (1.0f / 31.0f);
        const float mt = 1.0f - t;
        const float w0 = mt * mt * mt;
        const float w1 = 3.0f * mt * mt * t;
        const float w2 = 3.0f * mt * t * t;
        const float w3 = t * t * t;
        const float sx = w0 * p0x + w1 * p1x + w2 * p2x + w3 * p3x;
        const float sy = w0 * p0y + w1 * p1y + w2 * p2y + w3 * p3y;
        const float K  = SHARP_L2E_SQ;
        sLo[tid] = make_float2(-2.0f * K * sx, -2.0f * K * sy);
        sHi[tid] = make_float2(K, K * (sx * sx + sy * sy));
    }
    __syncthreads();

    // ---- Phase 2: each wave renders 16 pixels vs all 256 samples ----
    const int lane = tid & 31;
    const int m    = lane & 15;                 // pixel row within tile / sample col
    const bool hi  = lane >= 16;
    const int wave = tid >> 5;
    const int base = (blockIdx.x * WAVES_PER_BLOCK + wave) * PIX_PER_WAVE;

    // A operand (16x4 f32, ISA layout): lane L supplies row M = L&15.
    //   lanes 0..15 : {px, py}      (K-rows 0,1)
    //   lanes 16..31: {|p|^2, 1.0}  (K-rows 2,3)
    const float2 p = ((const float2*)pixel_grid)[base + m];
    const float pp = p.x * p.x + p.y * p.y;
    v2f a;
    a.x = hi ? pp   : p.x;
    a.y = hi ? 1.0f : p.y;

    // Preload ALL B-operand tiles into registers: paired ds_load_2addr_b64,
    // single dscnt wait, zero LDS stalls inside the WMMA loop.
    const float2* sb = hi ? sHi : sLo;
    v2f bs[N_TILES];
#pragma unroll
    for (int t = 0; t < N_TILES; ++t) {
        const float2 bv = sb[t * 16 + m];
        bs[t].x = bv.x;
        bs[t].y = bv.y;
    }

    float acc[8];
#pragma unroll
    for (int r = 0; r < 8; ++r) acc[r] = 0.0f;

#pragma unroll
    for (int t = 0; t < N_TILES; ++t) {
        v8f c = {};
        // D[16x16] = K * dist^2 tile: A(16x4 pixels) x B_K(4x16 samples)
        v8f d = __builtin_amdgcn_wmma_f32_16x16x4_f32(
            /*neg_a=*/false, a, /*neg_b=*/false, bs[t],
            /*c_mod=*/(short)0, c, /*reuse_a=*/false, /*reuse_b=*/false);

        // Stage 1: e[r] = sqrt(|K*d2|) -- abs folds as VOP3 src modifier on
        // v_sqrt_f32 (guards tiny negative cancellation); 8 independent chains.
        float e[8];
#pragma unroll
        for (int r = 0; r < 8; ++r)
            e[r] = __builtin_amdgcn_sqrtf(__builtin_fabsf(d[r]));

        // Stage 2: acc[r] += exp2(-e[r]) -- neg folds as src modifier on
        // v_exp_f32; 8 independent exp->add chains.
#pragma unroll
        for (int r = 0; r < 8; ++r)
            acc[r] += __builtin_amdgcn_exp2f(-e[r]);
    }

    // Reduce exp-sums across the 16 sample-columns (lanes 0..15 hold rows 0..7,
    // lanes 16..31 hold rows 8..15; xor masks <16 stay within each half).
#pragma unroll
    for (int mask = 1; mask < 16; mask <<= 1) {
#pragma unroll
        for (int r = 0; r < 8; ++r)
            acc[r] += __shfl_xor(acc[r], mask, 32);
    }

    // Lanes 0 and 16 finalize and store their half's 8 pixel rows.
    //   S   = sum_n exp(-256*dist_n)        (unchanged by the K-scaling)
    //   md  = -ln(S)/256
    //   arg = (STROKE_W - md)*(8/STROKE_W) = 8 + (200/256)*ln(S)
    //   out = 1 - sigmoid(arg) = 1/(1+exp2(K0 + K1*log2(S)))
    if (m == 0) {
        const int rowbase = base + (hi ? 8 : 0);
        const float K0 = 8.0f * 1.4426950408889634f;          // 8*log2(e)
        const float K1 = 0.78125f;                            // (200/256)*ln2*log2e
#pragma unroll
        for (int r = 0; r < 8; ++r) {
            const float l2S = __builtin_amdgcn_logf(acc[r]);  // v_log_f32 = log2
            const float g   = K0 + K1 * l2S;
            out[rowbase + r] = 1.0f / (1.0f + __builtin_amdgcn_exp2f(g));
        }
    }
}

extern "C" void kernel_launch(void* const* d_in, const int* in_sizes, int n_in,
                              void* d_out, int out_size, void* d_ws, size_t ws_size,
                              hipStream_t stream) {
    const float* cp = (const float*)d_in[0];   // control_points (8,4,2)
    const float* pg = (const float*)d_in[1];   // pixel_grid (262144,2)
    float* out = (float*)d_out;                // (1,512,512) float32

    const int n_pix  = IMG_SIZE * IMG_SIZE;
    const int blocks = n_pix / (PIX_PER_WAVE * WAVES_PER_BLOCK); // 2048
    bezier_glyph_wmma<<<blocks, THREADS, 0, stream>>>(cp, pg, out);
}